// RNN_86354612453407
// MI455X (gfx1250) — compile-verified
//
#include <hip/hip_runtime.h>
#include <math.h>

// ---------------------------------------------------------------------------
// 2-layer tanh RNN for MI455X (gfx1250, wave32, WMMA).
//   1) P = X@Wx + b  : one big parallel fp32-WMMA GEMM over all timesteps.
//   2) scan          : ONE persistent kernel per layer; each workgroup pins a
//                      1024x16 strip of Wh in 64KB LDS, runs all 512 steps with
//                      a device-wide atomic barrier between steps.
// Latency-bound problem (mem floor ~22us) -> kill launch overhead + weight
// refetch; keep full fp32 precision (V_WMMA_F32_16X16X4_F32).
// ---------------------------------------------------------------------------

typedef __attribute__((ext_vector_type(2))) float v2f;
typedef __attribute__((ext_vector_type(8))) float v8f;

constexpr int BB  = 64;      // batch
constexpr int TT  = 512;     // seq len
constexpr int HH  = 1024;    // hidden
constexpr int BH  = BB * HH;
constexpr long BTH = (long)BB * TT * HH;

__device__ __forceinline__ v8f wmma_f32(v2f a, v2f b, v8f c) {
    // D(16x16,f32) = A(16x4,f32) * B(4x16,f32) + C
    return __builtin_amdgcn_wmma_f32_16x16x4_f32(
        false, a, false, b, (short)0, c, false, false);
}

// ---------------------------------------------------------------------------
// Big GEMM: out[m, n] = sum_k A[m,k] * W[k,n] + bias[n]
//   M = TT*BB rows (m = t*BB + b), N = K = HH.
//   PERM=true : A row m comes from inputs laid out (B,T,H)  -> &X[(b*TT+t)*HH]
//   PERM=false: A row m is contiguous (T,B,H)               -> &X[m*HH]
// Block: 128 threads (4 waves). Block tile 64x64, wave tile 16x64.
// ---------------------------------------------------------------------------
template <bool PERM>
__global__ __launch_bounds__(128) void gemm_bias_kernel(
    const float* __restrict__ X, const float* __restrict__ W,
    const float* __restrict__ bias, float* __restrict__ out)
{
    __shared__ float As[64][17];   // padded: conflict-free column reads
    __shared__ float Bs[16][65];   // padded: conflict-free half-wave reads

    const int tid    = threadIdx.x;
    const int lane   = tid & 31;
    const int wave   = tid >> 5;            // 0..3 -> 16-row strip
    const int mBlock = blockIdx.x * 64;
    const int nBlock = blockIdx.y * 64;

    const int arow  = tid >> 1;             // 0..63
    const int akoff = (tid & 1) * 8;        // 0 or 8
    const int mG    = mBlock + arow;
    const float* aptr;
    if (PERM) {
        const int t = mG >> 6;              // /BB (BB==64)
        const int b = mG & 63;
        aptr = X + ((size_t)b * TT + t) * HH;
    } else {
        aptr = X + (size_t)mG * HH;
    }
    const int brow = tid >> 3;              // 0..15
    const int bcol = (tid & 7) * 8;         // 0..56

    const int lm = lane & 15;               // M (for A) / N (for B) in tile
    const int lk = (lane >> 4) << 1;        // K offset: 0 or 2

    v8f c0 = {}, c1 = {}, c2 = {}, c3 = {};

    for (int kk = 0; kk < HH; kk += 16) {
        { // stage A 64x16
            const float4* p = reinterpret_cast<const float4*>(aptr + kk + akoff);
            float4 v0 = p[0], v1 = p[1];
            float* d = &As[arow][akoff];
            d[0]=v0.x; d[1]=v0.y; d[2]=v0.z; d[3]=v0.w;
            d[4]=v1.x; d[5]=v1.y; d[6]=v1.z; d[7]=v1.w;
        }
        { // stage B 16x64
            const float4* p = reinterpret_cast<const float4*>(
                W + (size_t)(kk + brow) * HH + nBlock + bcol);
            float4 v0 = p[0], v1 = p[1];
            float* d = &Bs[brow][bcol];
            d[0]=v0.x; d[1]=v0.y; d[2]=v0.z; d[3]=v0.w;
            d[4]=v1.x; d[5]=v1.y; d[6]=v1.z; d[7]=v1.w;
        }
        __syncthreads();
#pragma unroll
        for (int k4 = 0; k4 < 16; k4 += 4) {
            const int ka = k4 + lk;
            v2f a;
            a.x = As[wave * 16 + lm][ka];
            a.y = As[wave * 16 + lm][ka + 1];
            v2f b0, b1, b2, b3;
            b0.x = Bs[ka][ 0 + lm]; b0.y = Bs[ka + 1][ 0 + lm];
            b1.x = Bs[ka][16 + lm]; b1.y = Bs[ka + 1][16 + lm];
            b2.x = Bs[ka][32 + lm]; b2.y = Bs[ka + 1][32 + lm];
            b3.x = Bs[ka][48 + lm]; b3.y = Bs[ka + 1][48 + lm];
            c0 = wmma_f32(a, b0, c0);
            c1 = wmma_f32(a, b1, c1);
            c2 = wmma_f32(a, b2, c2);
            c3 = wmma_f32(a, b3, c3);
        }
        __syncthreads();
    }

#pragma unroll
    for (int r = 0; r < 8; ++r) {
        const int ml = r + ((lane >> 4) << 3);          // 0..15 tile row
        const int m  = mBlock + wave * 16 + ml;
        float* orow  = out + (size_t)m * HH + nBlock;
        orow[ 0 + lm] = c0[r] + bias[nBlock +  0 + lm];
        orow[16 + lm] = c1[r] + bias[nBlock + 16 + lm];
        orow[32 + lm] = c2[r] + bias[nBlock + 32 + lm];
        orow[48 + lm] = c3[r] + bias[nBlock + 48 + lm];
    }
}

// ---------------------------------------------------------------------------
// Persistent scan kernel: runs ALL T steps of one layer.
// Grid: 64 blocks (one 16-col strip of Wh each), 128 threads = 4 waves,
// wave w handles rows [16w,16w+16) of the 64-row batch, full K=1024.
// Wh strip (1024x16 fp32 = 64KB) pinned in LDS for the whole scan.
// Device-wide barrier between steps via agent-scope atomics.
// ---------------------------------------------------------------------------
__global__ __launch_bounds__(128) void rnn_scan_kernel(
    const float* __restrict__ P,        // (T,B,H) preactivations incl. bias
    const float* __restrict__ Wh,       // (H,H)
    const float* __restrict__ h0,       // (B,H) initial hidden, row stride HH
    float* __restrict__ ys,             // ys(t) base = ys + t*tStride
    long tStride, long rStride,         // row stride within one timestep slab
    float* __restrict__ hfin,           // (B,H) final hidden state out
    unsigned int* __restrict__ barCtr)  // zeroed before launch
{
    __shared__ float Ws[HH][16];        // 64KB; conflict-free for frag reads

    const int tid   = threadIdx.x;
    const int lane  = tid & 31;
    const int wave  = tid >> 5;          // 0..3
    const int nBase = blockIdx.x * 16;   // this block's 16-col strip
    const int lm    = lane & 15;
    const int lk    = (lane >> 4) << 1;  // 0 or 2
    const int rowT  = wave * 16;

    // ---- pin Wh strip in LDS (once) ----
    {
        const int tRow = tid >> 2;           // 0..31
        const int tCol = (tid & 3) * 4;      // 0,4,8,12
        for (int k = tRow; k < HH; k += 32) {
            const float4 v = *reinterpret_cast<const float4*>(
                Wh + (size_t)k * HH + nBase + tCol);
            Ws[k][tCol + 0] = v.x; Ws[k][tCol + 1] = v.y;
            Ws[k][tCol + 2] = v.z; Ws[k][tCol + 3] = v.w;
        }
    }
    __syncthreads();

    for (int t = 0; t < TT; ++t) {
        const float* hp;
        long hs;
        if (t == 0) { hp = h0; hs = (long)HH; }
        else        { hp = ys + (size_t)(t - 1) * tStride; hs = rStride; }
        const float* arow = hp + (size_t)(rowT + lm) * hs;

        v8f c = {};
#pragma unroll 8
        for (int kk = 0; kk < HH; kk += 4) {
            const int ka = kk + lk;
            v2f a = *reinterpret_cast<const v2f*>(arow + ka);  // 8B aligned
            v2f b;
            b.x = Ws[ka][lm];
            b.y = Ws[ka + 1][lm];
            c = wmma_f32(a, b, c);
        }

        const float* Pt = P + (size_t)t * (size_t)BH;
        float* dstB = ys + (size_t)t * tStride;
#pragma unroll
        for (int r = 0; r < 8; ++r) {
            const int ml = r + ((lane >> 4) << 3);
            const int m  = rowT + ml;
            const int n  = nBase + lm;
            const float v = tanhf(c[r] + Pt[(size_t)m * HH + n]);
            dstB[(size_t)m * rStride + n] = v;
            if (t == TT - 1) hfin[(size_t)m * HH + n] = v;
        }

        if (t + 1 < TT) {
            // ---- device-wide barrier between timesteps ----
            __threadfence();                 // flush this wave's stores
            __syncthreads();
#if defined(__has_builtin)
#if __has_builtin(__builtin_amdgcn_s_cluster_barrier)
            __builtin_amdgcn_s_cluster_barrier();  // NOP when not clustered
#endif
#endif
            if (tid == 0) {
                __hip_atomic_fetch_add(barCtr, 1u, __ATOMIC_RELEASE,
                                       __HIP_MEMORY_SCOPE_AGENT);
                const unsigned target = (unsigned)gridDim.x * (unsigned)(t + 1);
                while (__hip_atomic_load(barCtr, __ATOMIC_ACQUIRE,
                                         __HIP_MEMORY_SCOPE_AGENT) < target) {
                    __builtin_amdgcn_s_sleep(2);
                }
            }
            __syncthreads();
            __threadfence();                 // refetch others' data
        }
    }
}

// ---------------------------------------------------------------------------
// Fallback per-step kernel (small-workspace path): fuses x@Wx + h@Wh.
// ---------------------------------------------------------------------------
__global__ __launch_bounds__(128) void rnn_step_kernel(
    const float* __restrict__ xrow, long xStride,
    const float* __restrict__ Wx,
    const float* __restrict__ bias,
    const float* __restrict__ hprev, long hStride,
    const float* __restrict__ Wh,
    float* __restrict__ dst, long dstStride)
{
    __shared__ float Bs[16][65];

    const int tid   = threadIdx.x;
    const int lane  = tid & 31;
    const int wave  = tid >> 5;
    const int mBase = blockIdx.x * 16;
    const int nBase = blockIdx.y * 64;
    const int lm    = lane & 15;
    const int lk    = (lane >> 4) << 1;
    const int brow  = tid >> 3;
    const int bcol  = (tid & 7) * 8;
    const int nTile = nBase + wave * 16;

    v8f c = {};
#pragma unroll 1
    for (int pass = 0; pass < 2; ++pass) {
        const float* W  = pass ? Wh : Wx;
        const float* Ar = pass ? hprev : xrow;
        const long   As = pass ? hStride : xStride;
        for (int kk = 0; kk < HH; kk += 16) {
            const float4* p = reinterpret_cast<const float4*>(
                W + (size_t)(kk + brow) * HH + nBase + bcol);
            float4 v0 = p[0], v1 = p[1];
            float* d = &Bs[brow][bcol];
            d[0]=v0.x; d[1]=v0.y; d[2]=v0.z; d[3]=v0.w;
            d[4]=v1.x; d[5]=v1.y; d[6]=v1.z; d[7]=v1.w;
            __syncthreads();
#pragma unroll
            for (int k4 = 0; k4 < 16; k4 += 4) {
                const int ka = k4 + lk;
                const float* ap = Ar + (size_t)(mBase + lm) * As + kk + ka;
                v2f a; a.x = ap[0]; a.y = ap[1];
                v2f b; b.x = Bs[ka][wave * 16 + lm]; b.y = Bs[ka + 1][wave * 16 + lm];
                c = wmma_f32(a, b, c);
            }
            __syncthreads();
        }
    }

#pragma unroll
    for (int r = 0; r < 8; ++r) {
        const int ml = r + ((lane >> 4) << 3);
        const int m  = mBase + ml;
        const int n  = nTile + lm;
        dst[(size_t)m * dstStride + n] = tanhf(c[r] + bias[n]);
    }
}

__global__ void copy_final_kernel(const float* __restrict__ src, long srcStride,
                                  float* __restrict__ dst)
{
    const int i = blockIdx.x * blockDim.x + threadIdx.x;
    if (i < BH) {
        const int b = i >> 10;
        const int h = i & 1023;
        dst[i] = src[(size_t)b * srcStride + h];
    }
}

extern "C" void kernel_launch(void* const* d_in, const int* in_sizes, int n_in,
                              void* d_out, int out_size, void* d_ws, size_t ws_size,
                              hipStream_t stream)
{
    const float* inputs = (const float*)d_in[0];   // (B,T,H)
    const float* h_prev = (const float*)d_in[1];   // (L,B,H)
    const float* W_xh   = (const float*)d_in[2];   // (L,H,H)
    const float* W_hh   = (const float*)d_in[3];   // (L,H,H)
    const float* b_h    = (const float*)d_in[4];   // (L,H)

    float* outs = (float*)d_out;                   // (B,T,H)
    float* hfin = outs + (size_t)BTH;              // (L,B,H)

    const size_t SZ = (size_t)TT * BB * HH;        // floats per (T,B,H) buffer
    const bool twoBuf = ws_size >= 2 * SZ * sizeof(float) + 64;
    float* P  = (float*)d_ws;                      // preactivations (T,B,H)
    float* Y0 = twoBuf ? P + SZ : (float*)d_ws;    // layer-0 ys (T,B,H)

    const dim3 blk(128);
    const dim3 gGemm(TT * BB / 64, HH / 64);       // (512, 16)
    const dim3 gScan(HH / 16);                     // 64 persistent blocks
    const dim3 gStep(BB / 16, HH / 64);
    const dim3 gCopy((BH + 255) / 256);
    const long THL = (long)TT * HH;

    if (twoBuf) {
        unsigned int* ctr = (unsigned int*)((char*)d_ws + 2 * SZ * sizeof(float));
        // ---- layer 0 ----
        gemm_bias_kernel<true><<<gGemm, blk, 0, stream>>>(inputs, W_xh, b_h, P);
        hipMemsetAsync(ctr, 0, sizeof(unsigned int), stream);
        rnn_scan_kernel<<<gScan, blk, 0, stream>>>(
            P, W_hh, h_prev, Y0, (long)BH, (long)HH, hfin, ctr);
        // ---- layer 1 (writes straight into (B,T,H) output) ----
        gemm_bias_kernel<false><<<gGemm, blk, 0, stream>>>(
            Y0, W_xh + (size_t)HH * HH, b_h + HH, P);
        hipMemsetAsync(ctr, 0, sizeof(unsigned int), stream);
        rnn_scan_kernel<<<gScan, blk, 0, stream>>>(
            P, W_hh + (size_t)HH * HH, h_prev + BH, outs, (long)HH, THL,
            hfin + BH, ctr);
    } else {
        // fallback: per-step fused kernels, one ws buffer
        for (int t = 0; t < TT; ++t) {
            const float* hp = (t == 0) ? h_prev : Y0 + (size_t)(t - 1) * BH;
            rnn_step_kernel<<<gStep, blk, 0, stream>>>(
                inputs + (size_t)t * HH, THL, W_xh, b_h,
                hp, (long)HH, W_hh, Y0 + (size_t)t * BH, (long)HH);
        }
        copy_final_kernel<<<gCopy, 256, 0, stream>>>(Y0 + (size_t)(TT - 1) * BH, (long)HH, hfin);
        for (int t = 0; t < TT; ++t) {
            const float* hp = (t == 0) ? h_prev + BH : outs + (size_t)(t - 1) * HH;
            const long hs   = (t == 0) ? (long)HH : THL;
            rnn_step_kernel<<<gStep, blk, 0, stream>>>(
                Y0 + (size_t)t * BH, (long)HH, W_xh + (size_t)HH * HH, b_h + HH,
                hp, hs, W_hh + (size_t)HH * HH, outs + (size_t)t * HH, THL);
        }
        copy_final_kernel<<<gCopy, 256, 0, stream>>>(outs + (size_t)(TT - 1) * HH, THL, hfin + BH);
    }
}